// GCNLayer_15144054685790
// MI455X (gfx1250) — compile-verified
//
#include <hip/hip_runtime.h>
#include <stdint.h>

typedef __attribute__((ext_vector_type(2))) float        v2f;
typedef __attribute__((ext_vector_type(8))) float        v8f;
typedef __attribute__((ext_vector_type(4))) unsigned int v4u;
typedef __attribute__((ext_vector_type(8))) int          v8i;
typedef __attribute__((ext_vector_type(4))) int          v4i;

#define N_NODES 50000
#define F_IN    256
#define F_OUT   64
#define M_TILES (N_NODES / 16)   // 3125

// ---------------------------------------------------------------------------
// Zero-init the output (harness poisons d_out with 0xAA before timing).
// ---------------------------------------------------------------------------
__global__ void gcn_zero_kernel(float* __restrict__ out, int n) {
    int i = blockIdx.x * blockDim.x + threadIdx.x;
    if (i < n) out[i] = 0.0f;
}

// ---------------------------------------------------------------------------
// Dense projection XW = X @ W with V_WMMA_F32_16X16X4_F32.
//
// One wave32 computes a full 16x64 output strip (4 accumulators), so each
// row of X is streamed from HBM exactly once (51.2 MB total instead of 4x).
// W (64 KB) is staged once per workgroup into LDS via the Tensor Data Mover
// (tensor_load_to_lds + s_wait_tensorcnt), falling back to a cooperative
// copy if the TDM builtin is unavailable. B fragments are then served by
// ds_load from LDS.
//
// Fragment layouts per ISA 7.12.2 (f32, wave32):
//   A 16x4 : lanes 0-15 M=lane, v0=K k0 / v1=K k0+1; lanes 16-31 K+=2
//   B 4x16 : lanes 0-15 N=lane, v0=K k0 / v1=K k0+1; lanes 16-31 K+=2
//   C/D    : VGPR r -> (M=r, N=lane) / (M=r+8, N=lane-16)
// ---------------------------------------------------------------------------
__global__ void __launch_bounds__(256)
gcn_gemm_wmma_kernel(const float* __restrict__ X,
                     const float* __restrict__ W,
                     float* __restrict__ XW) {
    __shared__ float w_lds[F_IN * F_OUT];   // 64 KB, row-major copy of W

#if defined(__HIP_DEVICE_COMPILE__) && __has_builtin(__builtin_amdgcn_tensor_load_to_lds)
    // --- TDM staging: one DMA descriptor, issued by wave 0 only ----------
    if (threadIdx.x < 32) {
        const uint64_t gaddr = (uint64_t)(uintptr_t)W;
        const uint32_t laddr = (uint32_t)(uintptr_t)&w_lds[0]; // LDS byte offset (low 32 bits)

        v4u g0; v8i g1;
        v4i g2 = {0, 0, 0, 0};
        v4i g3 = {0, 0, 0, 0};
        v8i g4 = {0, 0, 0, 0, 0, 0, 0, 0};
        // D# group0: [1:0]=count=1, [63:32]=lds_addr, [120:64]=global_addr,
        //            [127:126]=type=2
        g0[0] = 1u;
        g0[1] = laddr;
        g0[2] = (unsigned)(gaddr & 0xFFFFFFFFu);
        g0[3] = (unsigned)((gaddr >> 32) & 0x01FFFFFFu) | (2u << 30);
        // D# group1: data_size=2 (4B), tensor_dim0=64, tensor_dim1=256,
        //            tile_dim0=64, tile_dim1=256, tensor_dim0_stride=64
        g1[0] = (int)(2u << 16);        // [17:16] data_size; workgroup_mask=0
        g1[1] = (int)(64u << 16);       // tensor_dim0[15:0] @ bits 63:48
        g1[2] = (int)(256u << 16);      // tensor_dim1[15:0] @ bits 111:80 (low half)
        g1[3] = (int)(64u << 16);       // tile_dim0 @ bits 127:112
        g1[4] = (int)256;               // tile_dim1 @ bits 143:128 (tile_dim2=0)
        g1[5] = (int)64;                // tensor_dim0_stride[31:0]
        g1[6] = (int)(16384u << 16);    // tensor_dim1_stride[15:0] (ignored, 2D tile)
        g1[7] = 0;
        __builtin_amdgcn_tensor_load_to_lds(g0, g1, g2, g3, g4, /*cpol=*/0);
        __builtin_amdgcn_s_wait_tensorcnt(0);
    }
    __syncthreads();
#else
    // --- Fallback: cooperative vectorized copy ---------------------------
    for (int i = threadIdx.x; i < (F_IN * F_OUT) / 4; i += 256) {
        ((float4*)w_lds)[i] = ((const float4*)W)[i];
    }
    __syncthreads();
#endif

    const int lane   = threadIdx.x & 31;
    const int wave   = threadIdx.x >> 5;
    const int m_tile = blockIdx.x * 8 + wave;
    if (m_tile >= M_TILES) return;

    const int sub  = lane & 15;     // M (A fragment) / N (B fragment)
    const int kgrp = lane >> 4;     // selects K pair within a K-step of 4

    const float* __restrict__ xrow = X + (size_t)(m_tile * 16 + sub) * F_IN + 2 * kgrp;

    v8f acc0 = {}, acc1 = {}, acc2 = {}, acc3 = {};

    #pragma unroll 1
    for (int kb = 0; kb < F_IN; kb += 64) {
        // prefetch the next 256B chunk of this lane's X row
        __builtin_prefetch(xrow + kb + 64, 0, 3);
        #pragma unroll
        for (int k0 = kb; k0 < kb + 64; k0 += 4) {
            const int kq = k0 + 2 * kgrp;
            v2f a;
            a.x = xrow[k0 + 0];
            a.y = xrow[k0 + 1];
            const float* wr0 = &w_lds[(kq + 0) * F_OUT + sub];
            const float* wr1 = &w_lds[(kq + 1) * F_OUT + sub];
            v2f b0 = { wr0[0],  wr1[0]  };
            v2f b1 = { wr0[16], wr1[16] };
            v2f b2 = { wr0[32], wr1[32] };
            v2f b3 = { wr0[48], wr1[48] };
            acc0 = __builtin_amdgcn_wmma_f32_16x16x4_f32(false, a, false, b0, (short)0, acc0, false, false);
            acc1 = __builtin_amdgcn_wmma_f32_16x16x4_f32(false, a, false, b1, (short)0, acc1, false, false);
            acc2 = __builtin_amdgcn_wmma_f32_16x16x4_f32(false, a, false, b2, (short)0, acc2, false, false);
            acc3 = __builtin_amdgcn_wmma_f32_16x16x4_f32(false, a, false, b3, (short)0, acc3, false, false);
        }
    }

    // Store the 16x64 strip: VGPR r -> rows (m0 + kgrp*8 + r), column sub + 16*t
    float* __restrict__ orow = XW + (size_t)(m_tile * 16 + kgrp * 8) * F_OUT + sub;
    #pragma unroll
    for (int r = 0; r < 8; ++r) {
        float* p = orow + (size_t)r * F_OUT;
        p[0]  = acc0[r];
        p[16] = acc1[r];
        p[32] = acc2[r];
        p[48] = acc3[r];
    }
}

// ---------------------------------------------------------------------------
// COO aggregation: out[edge_row[e]] += edge_vals[e] * XW[edge_col[e]]
// One wave32 per edge; each lane handles 2 of the 64 features (b64 gather),
// scatter via global_atomic_add_f32 (XW + out resident in 192 MB L2).
// ---------------------------------------------------------------------------
__global__ void __launch_bounds__(256)
gcn_scatter_kernel(const float* __restrict__ XW,
                   const int*   __restrict__ edge_row,
                   const int*   __restrict__ edge_col,
                   const float* __restrict__ edge_vals,
                   float*       __restrict__ out,
                   int n_edges) {
    const int lane = threadIdx.x & 31;
    const int e    = blockIdx.x * 8 + (threadIdx.x >> 5);
    if (e >= n_edges) return;

    const int   r = edge_row[e];
    const int   c = edge_col[e];
    const float v = edge_vals[e];

    const float2 m = *(const float2*)(XW + (size_t)c * F_OUT + 2 * lane);
    float* dst = out + (size_t)r * F_OUT + 2 * lane;
    atomicAdd(dst + 0, v * m.x);
    atomicAdd(dst + 1, v * m.y);
}

// ---------------------------------------------------------------------------
// Launch: inputs in setup_inputs() order: X, W, edge_row, edge_col, edge_vals
// d_ws holds XW (50000*64 floats = 12.8 MB).
// ---------------------------------------------------------------------------
extern "C" void kernel_launch(void* const* d_in, const int* in_sizes, int n_in,
                              void* d_out, int out_size, void* d_ws, size_t ws_size,
                              hipStream_t stream) {
    const float* X         = (const float*)d_in[0];
    const float* W         = (const float*)d_in[1];
    const int*   edge_row  = (const int*)d_in[2];
    const int*   edge_col  = (const int*)d_in[3];
    const float* edge_vals = (const float*)d_in[4];
    float*       out       = (float*)d_out;
    float*       XW        = (float*)d_ws;

    const int n_edges = in_sizes[2];

    // 1) zero output
    {
        int n = N_NODES * F_OUT;
        gcn_zero_kernel<<<(n + 255) / 256, 256, 0, stream>>>(out, n);
    }
    // 2) WMMA GEMM: XW = X @ W  (8 m-tiles per block, 16x64 strip per wave)
    {
        int blocks = (M_TILES + 7) / 8;   // 391
        gcn_gemm_wmma_kernel<<<blocks, 256, 0, stream>>>(X, W, XW);
    }
    // 3) gather-scale-scatter
    {
        int blocks = (n_edges + 7) / 8;
        gcn_scatter_kernel<<<blocks, 256, 0, stream>>>(XW, edge_row, edge_col,
                                                       edge_vals, out, n_edges);
    }
}